// RankAwareMargin_3135326126284
// MI455X (gfx1250) — compile-verified
//
#include <hip/hip_runtime.h>
#include <math.h>

#define B_N 4096
#define K_DIM 128
#define MARGIN_F 0.2f
#define LDA 68   // 64 + 4 pad (dwords) for transposed LDS panels

typedef __attribute__((ext_vector_type(2))) float v2f;
typedef __attribute__((ext_vector_type(8))) float v8f;

#if defined(__has_builtin)
#if __has_builtin(__builtin_amdgcn_wmma_f32_16x16x4_f32)
#define HAVE_WMMA_F32X4 1
#endif
#endif

// ---------------- K0: per-row sum of squares ----------------
__global__ __launch_bounds__(256) void ram_sumsq(const float* __restrict__ X,
                                                 float* __restrict__ sq) {
  const int lane = threadIdx.x & 31;
  const int wave = threadIdx.x >> 5;
  const int row  = blockIdx.x * 8 + wave;
  float4 v = *(const float4*)(X + (size_t)row * K_DIM + lane * 4);
  float s = v.x * v.x + v.y * v.y + v.z * v.z + v.w * v.w;
#pragma unroll
  for (int off = 16; off > 0; off >>= 1) s += __shfl_down(s, off, 32);
  if (lane == 0) sq[row] = s;
}

// ---------------- K1: WMMA gram + sim_hat epilogue ----------------
// Block computes a 64x64 tile of sim_hat. 8 waves; wave (wr,wc) owns rows
// [wr*16,+16) and two 16x16 col tiles at wc*32 and wc*32+16.
__global__ __launch_bounds__(256) void ram_gram_sim(const float* __restrict__ X,
                                                    const float* __restrict__ sq,
                                                    const int* __restrict__ labels,
                                                    float* __restrict__ sim) {
  extern __shared__ float lds[];
  float* AsT = lds;                    // [K_DIM][LDA] : AsT[k][r] = X[rowBase+r][k]
  float* BsT = lds + K_DIM * LDA;      // [K_DIM][LDA] : BsT[k][c] = X[colBase+c][k]

  const int t = threadIdx.x;
  const int rowBase = blockIdx.y * 64;
  const int colBase = blockIdx.x * 64;

  __builtin_prefetch(sq + rowBase, 0, 0);      // global_prefetch_b8
  __builtin_prefetch(labels + colBase, 0, 0);

  // Stage both 64x128 panels (transposed) into LDS.
#pragma unroll
  for (int i = 0; i < 8; ++i) {
    const int e  = t + i * 256;   // 0..2047
    const int r  = e >> 5;        // 32 float4 per row
    const int c4 = e & 31;
    const int c  = c4 * 4;
    float4 a = *(const float4*)(X + (size_t)(rowBase + r) * K_DIM + c);
    float4 b = *(const float4*)(X + (size_t)(colBase + r) * K_DIM + c);
    AsT[(c + 0) * LDA + r] = a.x; AsT[(c + 1) * LDA + r] = a.y;
    AsT[(c + 2) * LDA + r] = a.z; AsT[(c + 3) * LDA + r] = a.w;
    BsT[(c + 0) * LDA + r] = b.x; BsT[(c + 1) * LDA + r] = b.y;
    BsT[(c + 2) * LDA + r] = b.z; BsT[(c + 3) * LDA + r] = b.w;
  }
  __syncthreads();

  const int lane = t & 31;
  const int w    = t >> 5;
  const int wr   = w >> 1;         // 0..3
  const int wc   = w & 1;          // 0..1
  const int kh   = lane >> 4;      // K half-select (A/B frag) / M+8 select (C/D)
  const int nlo  = lane & 15;
  const int mIdx = wr * 16 + nlo;  // A fragment M index (lanes 0-15 == 16-31)
  const int n0   = wc * 32;

  v8f c0 = {};
  v8f c1 = {};

  for (int kk = 0; kk < K_DIM; kk += 4) {
#ifdef HAVE_WMMA_F32X4
    const int kq = kk + 2 * kh;
    v2f a, b0, b1;
    a.x  = AsT[kq * LDA + mIdx];
    a.y  = AsT[(kq + 1) * LDA + mIdx];
    b0.x = BsT[kq * LDA + n0 + nlo];
    b0.y = BsT[(kq + 1) * LDA + n0 + nlo];
    b1.x = BsT[kq * LDA + n0 + 16 + nlo];
    b1.y = BsT[(kq + 1) * LDA + n0 + 16 + nlo];
    // D = A(16x4) * B(4x16) + C, f32 WMMA
    c0 = __builtin_amdgcn_wmma_f32_16x16x4_f32(false, a, false, b0, (short)0, c0, false, false);
    c1 = __builtin_amdgcn_wmma_f32_16x16x4_f32(false, a, false, b1, (short)0, c1, false, false);
#else
    // Scalar fallback honoring the same C/D layout (M = v + 8*kh, N = nlo).
#pragma unroll
    for (int kx = 0; kx < 4; ++kx) {
      const float bv0 = BsT[(kk + kx) * LDA + n0 + nlo];
      const float bv1 = BsT[(kk + kx) * LDA + n0 + 16 + nlo];
#pragma unroll
      for (int v = 0; v < 8; ++v) {
        const float av = AsT[(kk + kx) * LDA + wr * 16 + v + 8 * kh];
        c0[v] += av * bv0;
        c1[v] += av * bv1;
      }
    }
#endif
  }

  // Epilogue: sim_hat = -dist + (1-match)*MARGIN
  const int gRowBase = rowBase + wr * 16 + 8 * kh;   // C/D: M = v + 8*kh
  const int gCol0 = colBase + n0 + nlo;
  const int gCol1 = gCol0 + 16;
  const int lab0 = labels[gCol0];
  const int lab1 = labels[gCol1];
  const float sqc0 = sq[gCol0];
  const float sqc1 = sq[gCol1];
#pragma unroll
  for (int v = 0; v < 8; ++v) {
    const int gr = gRowBase + v;
    const float sqr = sq[gr];
    const int labr = labels[gr];
    {
      const float d2 = fmaxf(sqr + sqc0 - 2.0f * c0[v], 0.0f);
      const float dist = (d2 > 0.0f) ? sqrtf(d2) : 0.0f;
      const float m = (labr == lab0) ? 1.0f : 0.0f;
      sim[(size_t)gr * B_N + gCol0] = -dist + (1.0f - m) * MARGIN_F;
    }
    {
      const float d2 = fmaxf(sqr + sqc1 - 2.0f * c1[v], 0.0f);
      const float dist = (d2 > 0.0f) ? sqrtf(d2) : 0.0f;
      const float m = (labr == lab1) ? 1.0f : 0.0f;
      sim[(size_t)gr * B_N + gCol1] = -dist + (1.0f - m) * MARGIN_F;
    }
  }
}

// ---------------- K2: per-row rank-aware loss ----------------
// rank(j) = 1 + #{j2 : sim[j2] > sim[j]  or (sim[j2]==sim[j] and j2 < j)}
// (exactly the stable double-argsort rank). All lanes scan the same uniform
// j2 -> conflict-free LDS broadcast; float4 reads = ds_load_b128 broadcast.
// Non-match entries only need "rank<=k?" -> early-exit the count at chunk
// granularity (monotone condition, so chunked break is exact for fp entries).
// fn ordering among fn entries equals rank ordering, so
// fnpos(j) = #{fn j2 : rank(j2) < rank(j)} replaces the masked sort; since
// non-match ranks stay 0 and k >= 1 (self-match), (r2 > k) alone isolates fn.
__global__ __launch_bounds__(256) void ram_rowloss(const float* __restrict__ sim,
                                                   const int* __restrict__ labels,
                                                   float* __restrict__ partials) {
  __shared__ __align__(16) float s_sim[B_N];            // 16 KB
  __shared__ __align__(16) unsigned short s_rank[B_N];  // 8 KB
  __shared__ unsigned char s_match[B_N];                // 4 KB
  __shared__ float s_red[256];
  __shared__ int s_k;
  __shared__ int s_fpnum;

  const int row = blockIdx.x;
  const int t = threadIdx.x;
  if (t == 0) { s_k = 0; s_fpnum = 0; }
  __syncthreads();

  const int myLab = labels[row];
  const float* srow = sim + (size_t)row * B_N;
  __builtin_prefetch(srow + t * 4, 0, 0);
  int kloc = 0;
  for (int j = t; j < B_N; j += 256) {
    const float v = srow[j];
    s_sim[j] = v;
    const int m = (labels[j] == myLab) ? 1 : 0;
    s_match[j] = (unsigned char)m;
    s_rank[j] = 0;
    kloc += m;
  }
  atomicAdd(&s_k, kloc);
  __syncthreads();
  const int k = s_k;

  // Pass 1: exact ranks for matches; chunked early-exit fp scan for non-matches.
  float fp_part = 0.0f;
  int fp_loc = 0;
  for (int j = t; j < B_N; j += 256) {
    const float s = s_sim[j];
    if (s_match[j]) {
      int cnt = 0;
      for (int j2 = 0; j2 < B_N; j2 += 4) {
        const float4 v4 = *(const float4*)&s_sim[j2];
        cnt += (int)((v4.x > s) | ((v4.x == s) & (j2 + 0 < j)));
        cnt += (int)((v4.y > s) | ((v4.y == s) & (j2 + 1 < j)));
        cnt += (int)((v4.z > s) | ((v4.z == s) & (j2 + 2 < j)));
        cnt += (int)((v4.w > s) | ((v4.w == s) & (j2 + 3 < j)));
      }
      s_rank[j] = (unsigned short)(cnt + 1);
    } else {
      int cnt = 0;
      for (int j2 = 0; j2 < B_N; j2 += 16) {
#pragma unroll
        for (int q = 0; q < 16; q += 4) {
          const float4 v4 = *(const float4*)&s_sim[j2 + q];
          cnt += (int)((v4.x > s) | ((v4.x == s) & (j2 + q + 0 < j)));
          cnt += (int)((v4.y > s) | ((v4.y == s) & (j2 + q + 1 < j)));
          cnt += (int)((v4.z > s) | ((v4.z == s) & (j2 + q + 2 < j)));
          cnt += (int)((v4.w > s) | ((v4.w == s) & (j2 + q + 3 < j)));
        }
        if (cnt >= k) break;                       // rank > k for sure
      }
      if (cnt < k) {                               // false positive, exact rank
        const int rank = cnt + 1;
        fp_loc += 1;
        fp_part += s * (0.5f + 0.5f * (float)(k - rank + 1) / (float)k);
      }
    }
  }
  if (fp_loc) atomicAdd(&s_fpnum, fp_loc);
  __syncthreads();
  const int fp_num = s_fpnum;

  // Pass 2: false negatives; select those with fn-position < fp_num.
  float fn_part = 0.0f;
  for (int j = t; j < B_N; j += 256) {
    if (s_match[j]) {
      const int r = (int)s_rank[j];
      if (r > k) {
        int pos = 0;
        for (int j2 = 0; j2 < B_N; j2 += 4) {
          const ushort4 r4 = *(const ushort4*)&s_rank[j2];
          pos += (int)(((int)r4.x > k) & ((int)r4.x < r));
          pos += (int)(((int)r4.y > k) & ((int)r4.y < r));
          pos += (int)(((int)r4.z > k) & ((int)r4.z < r));
          pos += (int)(((int)r4.w > k) & ((int)r4.w < r));
        }
        if (pos < fp_num) {
          fn_part += s_sim[j] * (0.5f + 0.5f * (float)(r - k) / (float)(B_N - k));
        }
      }
    }
  }

  // Deterministic block reduction (no float atomics).
  s_red[t] = fp_part - fn_part;
  __syncthreads();
  for (int off = 128; off > 0; off >>= 1) {
    if (t < off) s_red[t] += s_red[t + off];
    __syncthreads();
  }
  if (t == 0) partials[row] = s_red[0];
}

// ---------------- K3: final reduction ----------------
__global__ __launch_bounds__(256) void ram_reduce(const float* __restrict__ partials,
                                                  float* __restrict__ out) {
  __shared__ float red[256];
  float s = 0.0f;
  for (int i = threadIdx.x; i < B_N; i += 256) s += partials[i];
  red[threadIdx.x] = s;
  __syncthreads();
  for (int off = 128; off > 0; off >>= 1) {
    if (threadIdx.x < off) red[threadIdx.x] += red[threadIdx.x + off];
    __syncthreads();
  }
  if (threadIdx.x == 0) out[0] = red[0];
}

extern "C" void kernel_launch(void* const* d_in, const int* in_sizes, int n_in,
                              void* d_out, int out_size, void* d_ws, size_t ws_size,
                              hipStream_t stream) {
  const float* X      = (const float*)d_in[0];   // [4096,128] f32
  const int*   labels = (const int*)d_in[1];     // [4096] i32
  float* out = (float*)d_out;                    // scalar f32

  // Workspace layout: sq[4096] | partials[4096] | sim[4096*4096]
  float* sq       = (float*)d_ws;
  float* partials = sq + B_N;
  float* sim      = partials + B_N;

  ram_sumsq<<<dim3(B_N / 8), dim3(256), 0, stream>>>(X, sq);

  const size_t gramLds = (size_t)2 * K_DIM * LDA * sizeof(float); // 69632 B
  ram_gram_sim<<<dim3(B_N / 64, B_N / 64), dim3(256), gramLds, stream>>>(X, sq, labels, sim);

  ram_rowloss<<<dim3(B_N), dim3(256), 0, stream>>>(sim, labels, partials);

  ram_reduce<<<dim3(1), dim3(256), 0, stream>>>(partials, out);
}